// FerroelectricBasisConv2d_87978110091302
// MI455X (gfx1250) — compile-verified
//
#include <hip/hip_runtime.h>
#include <math.h>

// FerroelectricBasisConv2d for MI455X (gfx1250, wave32).
// Not a GEMM: weights live inside tanh() with the activation, so WMMA cannot
// apply. CDNA5 features used instead: native v_tanh_f32 (TRANS pipe), async
// global->LDS copies (ASYNCcnt / s_wait_asynccnt), wave32 lane->pixel mapping
// for coalescing + LDS broadcast of wave-uniform params.
//
// NOTE: fbc_main is defined FIRST so the disasm snippet shows its codegen.

#ifndef __has_builtin
#define __has_builtin(x) 0
#endif

#define AS1 __attribute__((address_space(1)))
#define AS3 __attribute__((address_space(3)))

typedef int v2i __attribute__((vector_size(2 * sizeof(int))));
typedef int v4i __attribute__((vector_size(4 * sizeof(int))));

// ---- native tanh (gfx1250 TRANS op) with fallback ----
#if __has_builtin(__builtin_amdgcn_tanhf)
__device__ __forceinline__ float fast_tanh(float x) { return __builtin_amdgcn_tanhf(x); }
#elif __has_builtin(__builtin_amdgcn_tanh_f32)
__device__ __forceinline__ float fast_tanh(float x) { return __builtin_amdgcn_tanh_f32(x); }
#else
__device__ __forceinline__ float fast_tanh(float x) { return tanhf(x); }
#endif

// ---- async global->LDS staging (gfx1250) with fallback ----
#if __has_builtin(__builtin_amdgcn_global_load_async_to_lds_b64) && \
    __has_builtin(__builtin_amdgcn_global_load_async_to_lds_b128) && \
    __has_builtin(__builtin_amdgcn_s_wait_asynccnt)
#define HAVE_ASYNC 1
#else
#define HAVE_ASYNC 0
#endif

__device__ __forceinline__ void cp_lds_b128(float* lds_dst, const float* gsrc) {
#if HAVE_ASYNC
  __builtin_amdgcn_global_load_async_to_lds_b128((AS1 v4i*)gsrc, (AS3 v4i*)lds_dst, 0, 0);
#else
  *(float4*)lds_dst = *(const float4*)gsrc;
#endif
}

__device__ __forceinline__ void cp_lds_b64(float* lds_dst, const float* gsrc) {
#if HAVE_ASYNC
  __builtin_amdgcn_global_load_async_to_lds_b64((AS1 v2i*)gsrc, (AS3 v2i*)lds_dst, 0, 0);
#else
  *(float2*)lds_dst = *(const float2*)gsrc;
#endif
}

__device__ __forceinline__ void async_stage_wait() {
#if HAVE_ASYNC
  __builtin_amdgcn_s_wait_asynccnt(0);
#endif
}

// Problem constants
#define B_      4
#define CIN     16
#define H_      32
#define W_      32
#define COUT    32
#define RED     432          // Cin*K*kH*kW = 16*3*3*3
#define HP      34           // padded H
#define WP      34           // padded W
#define CO_PER_BLK 8

// ---------------- main kernel (defined first for disasm visibility) ----------------
// grid: (COUT/8, Hout, B); block: 256 = 8 waves; wave w -> co, lane -> wo.
__global__ __launch_bounds__(256)
void fbc_main(const float* __restrict__ kten, const float* __restrict__ Ec,
              const float* __restrict__ pc,   const float* __restrict__ cconst,
              const float* __restrict__ xpad, float* __restrict__ out) {
  __shared__ __align__(16) float xs[CIN * 3 * WP];        // 1632 floats: [ci][row 0..2][WP]
  __shared__ __align__(16) float ks[CO_PER_BLK * RED];    // 3456
  __shared__ __align__(16) float es[CO_PER_BLK * RED];    // 3456
  __shared__ __align__(16) float ps[CO_PER_BLK * RED];    // 3456 (Ps*coef)

  const int tid  = threadIdx.x;
  const int w    = tid >> 5;       // wave id 0..7
  const int lane = tid & 31;       // wo
  const int co_base = blockIdx.x * CO_PER_BLK;
  const int ho = blockIdx.y;
  const int b  = blockIdx.z;
  const int co = co_base + w;

  // --- stage 8 co's worth of params: 3456 floats each = 864 x b128 ---
  {
    const float* gk = kten + (size_t)co_base * RED;
    const float* ge = Ec   + (size_t)co_base * RED;
    const float* gp = pc   + (size_t)co_base * RED;
    for (int i = tid; i < (CO_PER_BLK * RED) / 4; i += 256) {
      cp_lds_b128(&ks[i * 4], gk + i * 4);
      cp_lds_b128(&es[i * 4], ge + i * 4);
      cp_lds_b128(&ps[i * 4], gp + i * 4);
    }
    // --- stage x tile: per ci, 3 padded rows (102 contiguous floats) = 51 x b64 ---
    const float* gx = xpad + ((size_t)(b * CIN) * HP + ho) * WP;   // ci=0 segment start
    for (int i = tid; i < CIN * 51; i += 256) {
      int ci = i / 51;
      int r  = i - ci * 51;
      cp_lds_b64(&xs[ci * 102 + r * 2], gx + (size_t)ci * (HP * WP) + r * 2);
    }
  }
  async_stage_wait();
  __syncthreads();

  // --- compute: 432-term reduction, xe hoisted over kk, 3 accumulators ---
  const float* kw = &ks[w * RED];
  const float* ew = &es[w * RED];
  const float* pw = &ps[w * RED];
  float acc0 = 0.0f, acc1 = 0.0f, acc2 = 0.0f;

  #pragma unroll 1
  for (int ci = 0; ci < CIN; ++ci) {
    const float* xrow = &xs[ci * 102];
    const int mc = ci * 27;
    #pragma unroll
    for (int i = 0; i < 3; ++i) {
      #pragma unroll
      for (int j = 0; j < 3; ++j) {
        const float xe = xrow[i * WP + lane + j];
        const int mb = mc + i * 3 + j;            // flat m = ci*27 + kk*9 + i*3 + j
        {
          const float e  = ew[mb];
          const float bm = fmaf(0.1f, fast_tanh(5.0f * (xe + e)), 0.9f);
          const float t2 = fast_tanh(kw[mb] * fmaf(e, bm, xe));
          acc0 = fmaf(pw[mb], t2, acc0);
        }
        {
          const float e  = ew[mb + 9];
          const float bm = fmaf(0.1f, fast_tanh(5.0f * (xe + e)), 0.9f);
          const float t2 = fast_tanh(kw[mb + 9] * fmaf(e, bm, xe));
          acc1 = fmaf(pw[mb + 9], t2, acc1);
        }
        {
          const float e  = ew[mb + 18];
          const float bm = fmaf(0.1f, fast_tanh(5.0f * (xe + e)), 0.9f);
          const float t2 = fast_tanh(kw[mb + 18] * fmaf(e, bm, xe));
          acc2 = fmaf(pw[mb + 18], t2, acc2);
        }
      }
    }
  }

  const float r = (acc0 + acc1) + (acc2 + cconst[co]);
  out[(((size_t)b * COUT + co) * H_ + ho) * W_ + lane] = r;   // [B][Cout][H][W], lane=wo coalesced
}

// ---------------- prep kernels ----------------
__global__ void fbc_prep_pc(const float* __restrict__ Ps, const float* __restrict__ coef,
                            float* __restrict__ pc, int n) {
  int i = blockIdx.x * blockDim.x + threadIdx.x;
  if (i < n) pc[i] = Ps[i] * coef[i];
}

__global__ void fbc_pad_x(const float* __restrict__ x, float* __restrict__ xpad) {
  // xpad layout: [B][CIN][HP][WP], zero halo of 1
  int i = blockIdx.x * blockDim.x + threadIdx.x;
  const int total = B_ * CIN * HP * WP;
  if (i >= total) return;
  int c34 = i % WP;
  int t   = i / WP;
  int r34 = t % HP;
  t /= HP;                       // t = b*CIN + ci
  int r = r34 - 1, c = c34 - 1;
  float v = 0.0f;
  if (r >= 0 && r < H_ && c >= 0 && c < W_) v = x[(t * H_ + r) * W_ + c];
  xpad[i] = v;
}

__global__ __launch_bounds__(256)
void fbc_prep_const(const float* __restrict__ bias, const float* __restrict__ coef,
                    const float* __restrict__ out_bias, float* __restrict__ cconst) {
  __shared__ float red[256];
  const int co = blockIdx.x, tid = threadIdx.x;
  float s = 0.0f;
  for (int m = tid; m < RED; m += 256) s += bias[co * RED + m] * coef[co * RED + m];
  red[tid] = s;
  __syncthreads();
  for (int st = 128; st > 0; st >>= 1) {
    if (tid < st) red[tid] += red[tid + st];
    __syncthreads();
  }
  if (tid == 0) cconst[co] = red[0] + out_bias[co];
}

// ---------------- launch ----------------
extern "C" void kernel_launch(void* const* d_in, const int* in_sizes, int n_in,
                              void* d_out, int out_size, void* d_ws, size_t ws_size,
                              hipStream_t stream) {
  (void)in_sizes; (void)n_in; (void)out_size; (void)ws_size;
  const float* x        = (const float*)d_in[0];
  const float* k        = (const float*)d_in[1];
  const float* Ec       = (const float*)d_in[2];
  const float* Ps       = (const float*)d_in[3];
  const float* bias     = (const float*)d_in[4];
  const float* coef     = (const float*)d_in[5];
  const float* out_bias = (const float*)d_in[6];
  float* out = (float*)d_out;

  // workspace layout (floats): pc[13824] | cconst[32] | xpad[4*16*34*34]
  float* wsf    = (float*)d_ws;
  float* pc     = wsf;
  float* cconst = wsf + (COUT * RED);
  float* xpad   = wsf + (COUT * RED) + COUT;   // 16B-aligned (13856*4 bytes)

  const int npc = COUT * RED;                  // 13824
  fbc_prep_pc<<<(npc + 255) / 256, 256, 0, stream>>>(Ps, coef, pc, npc);
  const int npad = B_ * CIN * HP * WP;         // 73984
  fbc_pad_x<<<(npad + 255) / 256, 256, 0, stream>>>(x, xpad);
  fbc_prep_const<<<COUT, 256, 0, stream>>>(bias, coef, out_bias, cconst);

  dim3 grid(COUT / CO_PER_BLK, H_, B_);        // (4, 32, 4)
  fbc_main<<<grid, 256, 0, stream>>>(k, Ec, pc, cconst, xpad, out);
}